// PointStretchLoss_8117488189444
// MI455X (gfx1250) — compile-verified
//
#include <hip/hip_runtime.h>
#include <math.h>

typedef __attribute__((ext_vector_type(2))) float v2f;
typedef __attribute__((ext_vector_type(8))) float v8f;

#define KNN 16
#define TSTRIDE 36   // words; 36*m mod 64 spaced by 4 -> conflict-free b128 column reads

__global__ void PS_init_out(float* out) { out[0] = 0.0f; }

// Branchless sorted-insert into a 16-entry ascending register list (v_cndmask network).
__device__ __forceinline__ void insert16(float (&bd)[KNN], int (&bi)[KNN], float v, int j) {
#pragma unroll
    for (int t = 0; t < KNN; ++t) {
        const bool  c  = v < bd[t];
        const float nb = c ? v : bd[t];
        const int   ni = c ? j : bi[t];
        v = c ? bd[t] : v;
        j = c ? bi[t] : j;
        bd[t] = nb;
        bi[t] = ni;
    }
}

// One wave32 per 16-row query tile. Two V_WMMA_F32_16X16X4_F32 per iteration
// compute a 16x32 tile of d^2 directly: A is pre-scaled by -2, and the C
// accumulator is seeded with sqa[M] + sqb, so D = sqa + sqb - 2*dot.
__global__ __launch_bounds__(32) void PS_stretch_kernel(
    const float* __restrict__ pref, const float* __restrict__ pts,
    float* __restrict__ out, int N, float scale)
{
    __shared__ float sqa[16];
    __shared__ __align__(16) float tile[16 * TSTRIDE];   // 16 rows x 32 cols (+pad)
    __shared__ float md[32 * KNN];
    __shared__ int   mi[32 * KNN];

    const int lane = threadIdx.x;      // 0..31
    const int col  = lane & 15;
    const int hi   = lane >> 4;        // A/B layout half: K={0,1} vs K={2,3}
    const int nct  = N >> 5;           // column super-tiles of 32
    const int nrt  = N >> 4;
    const int b    = blockIdx.x / nrt;
    const int it   = blockIdx.x - b * nrt;
    const int row0 = it << 4;

    const float* __restrict__ refb = pref + (size_t)b * N * 3;
    const float* __restrict__ ptsb = pts  + (size_t)b * N * 3;

    // ---- A tile (16 query points) in WMMA 16x4 f32 A-layout, pre-scaled by -2 ----
    const int m = row0 + col;
    const float ax = refb[m * 3 + 0];
    const float ay = refb[m * 3 + 1];
    const float az = refb[m * 3 + 2];
    if (hi == 0) sqa[col] = ax * ax + ay * ay + az * az;
    v2f A;
    A.x = -2.0f * (hi ? az : ax);
    A.y = hi ? 0.0f : (-2.0f * ay);
    __syncthreads();

    // loop-invariant per-lane row norms for the C accumulator (rows M = r + 8*hi)
    float sqar[8];
#pragma unroll
    for (int r = 0; r < 8; ++r) sqar[r] = sqa[r + (hi << 3)];

    // per-half-lane partial top-16 (sorted ascending)
    float bd[KNN];
    int   bi[KNN];
#pragma unroll
    for (int k = 0; k < KNN; ++k) { bd[k] = 3.0e38f; bi[k] = 0; }

    const int rl    = lane & 15;       // row handled in scan phase
    const int myrow = row0 + rl;
    const int cbase = hi << 4;         // column half handled in scan phase

    for (int jt = 0; jt < nct; ++jt) {
        const int j0 = (jt << 5) + col;        // candidate for B tile 0
        const int j1 = j0 + 16;                // candidate for B tile 1
        if (jt + 1 < nct)
            __builtin_prefetch(&refb[(size_t)(j0 + 32) * 3], 0, 1);

        const float b0x = refb[j0 * 3 + 0], b0y = refb[j0 * 3 + 1], b0z = refb[j0 * 3 + 2];
        const float b1x = refb[j1 * 3 + 0], b1y = refb[j1 * 3 + 1], b1z = refb[j1 * 3 + 2];
        const float sqb0 = b0x * b0x + b0y * b0y + b0z * b0z;
        const float sqb1 = b1x * b1x + b1y * b1y + b1z * b1z;

        v2f B0, B1;
        B0.x = hi ? b0z : b0x;  B0.y = hi ? 0.0f : b0y;
        B1.x = hi ? b1z : b1x;  B1.y = hi ? 0.0f : b1y;

        // seed accumulators with sqa[M] + sqb -> WMMA emits d^2 directly
        v8f acc0, acc1;
#pragma unroll
        for (int r = 0; r < 8; ++r) {
            acc0[r] = sqar[r] + sqb0;
            acc1[r] = sqar[r] + sqb1;
        }
        acc0 = __builtin_amdgcn_wmma_f32_16x16x4_f32(false, A, false, B0, (short)0, acc0, false, false);
        acc1 = __builtin_amdgcn_wmma_f32_16x16x4_f32(false, A, false, B1, (short)0, acc1, false, false);

        // D layout: lane holds rows M = r + 8*hi, column = col
#pragma unroll
        for (int r = 0; r < 8; ++r) {
            const int M = r + (hi << 3);
            tile[M * TSTRIDE + col]      = acc0[r];
            tile[M * TSTRIDE + 16 + col] = acc1[r];
        }
        __syncthreads();

        // ---- scan: 16 candidates per lane, float4 loads, min4 pre-filter ----
        const float* rowp = &tile[rl * TSTRIDE + cbase];
        const int gbase = (jt << 5) + cbase;
#pragma unroll
        for (int c4 = 0; c4 < 4; ++c4) {
            const float4 q = *(const float4*)(rowp + (c4 << 2));
            const float m4 = fminf(fminf(q.x, q.y), fminf(q.z, q.w));
            if (m4 < bd[KNN - 1]) {
                const int g = gbase + (c4 << 2);
                if (q.x < bd[KNN - 1] && (g + 0) != myrow) insert16(bd, bi, q.x, g + 0);
                if (q.y < bd[KNN - 1] && (g + 1) != myrow) insert16(bd, bi, q.y, g + 1);
                if (q.z < bd[KNN - 1] && (g + 2) != myrow) insert16(bd, bi, q.z, g + 2);
                if (q.w < bd[KNN - 1] && (g + 3) != myrow) insert16(bd, bi, q.w, g + 3);
            }
        }
        __syncthreads();
    }

    // ---- merge the two per-row partial lists (lane rl and rl+16) ----
#pragma unroll
    for (int k = 0; k < KNN; ++k) {
        md[lane * KNN + k] = bd[k];
        mi[lane * KNN + k] = bi[k];
    }
    __syncthreads();

    float sum = 0.0f;
    if (lane < 16) {
        int ia = 0, ib = 0;
        float rd[KNN]; int ri[KNN];
#pragma unroll
        for (int k = 0; k < KNN; ++k) {
            const float da = md[lane * KNN + ia];
            const float db = md[(lane + 16) * KNN + ib];
            const bool t = da <= db;
            rd[k] = t ? da : db;
            ri[k] = t ? mi[lane * KNN + ia] : mi[(lane + 16) * KNN + ib];
            ia += t ? 1 : 0;
            ib += t ? 0 : 1;
        }

        // ---- stretch = mean(max(dist/dist_ref - 1, 0)) ----
        const float px = ptsb[myrow * 3 + 0];
        const float py = ptsb[myrow * 3 + 1];
        const float pz = ptsb[myrow * 3 + 2];
#pragma unroll
        for (int k = 0; k < KNN; ++k) {
            const int jn = ri[k];
            const float dref = sqrtf(fmaxf(rd[k], 0.0f));
            const float dx = ptsb[jn * 3 + 0] - px;
            const float dy = ptsb[jn * 3 + 1] - py;
            const float dz = ptsb[jn * 3 + 2] - pz;
            const float d  = sqrtf(dx * dx + dy * dy + dz * dz);
            sum += fmaxf(d / dref - 1.0f, 0.0f);
        }
    }

    // block-level reduction -> single atomic
    __syncthreads();
    md[lane] = sum;
    __syncthreads();
    if (lane == 0) {
        float tot = 0.0f;
#pragma unroll
        for (int k = 0; k < 16; ++k) tot += md[k];
        atomicAdd(out, tot * scale);
    }
}

extern "C" void kernel_launch(void* const* d_in, const int* in_sizes, int n_in,
                              void* d_out, int out_size, void* d_ws, size_t ws_size,
                              hipStream_t stream) {
    const float* pref = (const float*)d_in[0];   // points_ref (B,N,3) f32
    const float* pts  = (const float*)d_in[1];   // points     (B,N,3) f32
    float* out = (float*)d_out;                  // scalar mean

    const int B = 4;
    const int N = in_sizes[0] / (B * 3);         // 8192
    const float scale = 1.0f / ((float)B * (float)N * (float)KNN);

    PS_init_out<<<1, 1, 0, stream>>>(out);
    const int blocks = B * (N / 16);             // one wave32 per 16-row tile
    PS_stretch_kernel<<<blocks, 32, 0, stream>>>(pref, pts, out, N, scale);
}